// GraphConvolution_19413252178072
// MI455X (gfx1250) — compile-verified
//
#include <hip/hip_runtime.h>
#include <hip/hip_bf16.h>
#include <math.h>

typedef __attribute__((ext_vector_type(2))) float v2f;
typedef __attribute__((ext_vector_type(8))) float v8f;

#define NN 50000
#define NE 800000
#define DD 128

// ---------------------------------------------------------------------------
// Kernel 1: support = inputs @ weight via V_WMMA_F32_16X16X4_F32
// 256 threads = 8 waves; block computes 16 rows x 128 cols; wave w -> cols [16w,16w+16)
// ---------------------------------------------------------------------------
__global__ __launch_bounds__(256) void gemm_wmma_f32(const float* __restrict__ A,
                                                     const float* __restrict__ W,
                                                     float* __restrict__ S) {
  __shared__ float As[16 * 132];          // padded stride 132 -> conflict-free frag reads
  const int row0 = blockIdx.x * 16;
  const int tid  = threadIdx.x;

  // Cooperative, coalesced stage of the 16x128 A tile (512 float4, 2 per thread)
  {
    const float4* Ag = reinterpret_cast<const float4*>(A + (size_t)row0 * DD);
    #pragma unroll
    for (int i = tid; i < 512; i += 256) {
      int r  = i >> 5;                    // 32 float4 per row
      int c4 = i & 31;
      float4 v = Ag[r * 32 + c4];
      float* dst = &As[r * 132 + c4 * 4];
      dst[0] = v.x; dst[1] = v.y; dst[2] = v.z; dst[3] = v.w;
    }
  }
  __syncthreads();

  const int wave = tid >> 5;
  const int lane = tid & 31;
  const int col0 = wave * 16;
  const int n    = lane & 15;             // N for B/C frags, M for A frag
  const int koff = (lane >> 4) * 2;       // lane-half K offset

  v8f c = {};
  #pragma unroll 4
  for (int k = 0; k < DD; k += 4) {
    v2f a, b;
    const float* ap = &As[n * 132 + k + koff];
    a.x = ap[0];                          // (M=n, K=k+koff)
    a.y = ap[1];                          // (M=n, K=k+koff+1)
    b.x = W[(k + koff)     * DD + col0 + n];   // (K, N=col0+n)
    b.y = W[(k + koff + 1) * DD + col0 + n];
    c = __builtin_amdgcn_wmma_f32_16x16x4_f32(false, a, false, b,
                                              (short)0, c, false, false);
  }

  const int mbase = (lane >> 4) * 8;      // C/D: VGPR r -> M=r (+8 upper half)
  #pragma unroll
  for (int r = 0; r < 8; ++r) {
    S[(size_t)(row0 + mbase + r) * DD + col0 + n] = c[r];
  }
}

// ---------------------------------------------------------------------------
// Kernel 2: zero the accumulator (d_out used as agg buffer)
// ---------------------------------------------------------------------------
__global__ __launch_bounds__(256) void zero_f4(float4* __restrict__ p, int n4) {
  int i = blockIdx.x * blockDim.x + threadIdx.x;
  if (i < n4) p[i] = make_float4(0.f, 0.f, 0.f, 0.f);
}

// ---------------------------------------------------------------------------
// Kernel 3: COO SpMM scatter: agg[row] += val * support[col]
// One wave per edge; each lane handles 4 floats (float4 gather + 4 f32 atomics)
// ---------------------------------------------------------------------------
__global__ __launch_bounds__(256) void scatter_edges(const float* __restrict__ S,
                                                     const float* __restrict__ vals,
                                                     const int* __restrict__ rows,
                                                     const int* __restrict__ cols,
                                                     float* __restrict__ agg) {
  int e    = (blockIdx.x * blockDim.x + threadIdx.x) >> 5;
  int lane = threadIdx.x & 31;
  if (e >= NE) return;
  float v = vals[e];
  int   src = cols[e];
  int   dst = rows[e];
  float4 s = reinterpret_cast<const float4*>(S + (size_t)src * DD)[lane];
  float* o = agg + (size_t)dst * DD + lane * 4;
  atomicAdd(o + 0, s.x * v);              // global_atomic_add_f32 (no return)
  atomicAdd(o + 1, s.y * v);
  atomicAdd(o + 2, s.z * v);
  atomicAdd(o + 3, s.w * v);
}

// ---------------------------------------------------------------------------
// Kernel 4: out = elu(gate0*agg + gate1*inputs + bias), in place on d_out
// ---------------------------------------------------------------------------
__global__ __launch_bounds__(256) void finalize(float* __restrict__ out,      // holds agg
                                                const float* __restrict__ x,
                                                const float* __restrict__ bias,
                                                const float* __restrict__ alpha) {
  int i = blockIdx.x * blockDim.x + threadIdx.x;        // float4 index
  if (i >= NN * DD / 4) return;
  float a0 = alpha[0], a1 = alpha[1];
  float m  = fmaxf(a0, a1);
  float e0 = __expf(a0 - m), e1 = __expf(a1 - m);
  float inv = 1.f / (e0 + e1);
  float g0 = e0 * inv, g1 = e1 * inv;

  int d4 = i & (DD / 4 - 1);
  float4 ag = reinterpret_cast<float4*>(out)[i];
  float4 xi = reinterpret_cast<const float4*>(x)[i];
  float4 b  = reinterpret_cast<const float4*>(bias)[d4];
  float4 o;
  o.x = ag.x * g0 + xi.x * g1 + b.x;
  o.y = ag.y * g0 + xi.y * g1 + b.y;
  o.z = ag.z * g0 + xi.z * g1 + b.z;
  o.w = ag.w * g0 + xi.w * g1 + b.w;
  o.x = o.x > 0.f ? o.x : __expf(o.x) - 1.f;
  o.y = o.y > 0.f ? o.y : __expf(o.y) - 1.f;
  o.z = o.z > 0.f ? o.z : __expf(o.z) - 1.f;
  o.w = o.w > 0.f ? o.w : __expf(o.w) - 1.f;
  reinterpret_cast<float4*>(out)[i] = o;
}

// ---------------------------------------------------------------------------
extern "C" void kernel_launch(void* const* d_in, const int* in_sizes, int n_in,
                              void* d_out, int out_size, void* d_ws, size_t ws_size,
                              hipStream_t stream) {
  const float* inputs   = (const float*)d_in[0];   // [NN, DD]
  const float* weight   = (const float*)d_in[1];   // [DD, DD]
  const float* bias     = (const float*)d_in[2];   // [DD]
  const float* alpha    = (const float*)d_in[3];   // [2]
  const float* adj_vals = (const float*)d_in[4];   // [NE]
  const int*   adj_rows = (const int*)d_in[5];     // [NE]
  const int*   adj_cols = (const int*)d_in[6];     // [NE]

  float* support = (float*)d_ws;                   // [NN*DD] = 25.6 MB scratch
  float* out     = (float*)d_out;                  // used as agg accumulator, then final

  // 1) zero accumulator (d_out)
  {
    int n4 = NN * DD / 4;
    zero_f4<<<(n4 + 255) / 256, 256, 0, stream>>>((float4*)out, n4);
  }
  // 2) support = inputs @ weight (WMMA fp32)
  gemm_wmma_f32<<<NN / 16, 256, 0, stream>>>(inputs, weight, support);
  // 3) scatter-add over edges (8 edges per 256-thread block)
  scatter_edges<<<(NE + 7) / 8, 256, 0, stream>>>(support, adj_vals, adj_rows,
                                                  adj_cols, out);
  // 4) gated combine + bias + ELU, in place
  {
    int n4 = NN * DD / 4;
    finalize<<<(n4 + 255) / 256, 256, 0, stream>>>(out, inputs, bias, alpha);
  }
}